// StaticDifferentiableLQR_2697239462037
// MI455X (gfx1250) — compile-verified
//
#include <hip/hip_runtime.h>
#include <math.h>

// LQR Riccati recursion, 1M independent 3x3 problems, HORIZON=20.
// Compute-bound on FP32 VALU (~1.4 KFLOP/elem vs 28 B/elem -> ~50 FLOP/B,
// far above the 23.3 TB/s roofline knee). One thread per element, P kept
// symmetric in 6 registers, B sparsity + A structure folded in, 2x2 solve
// done analytically. Input tile staged to LDS via the gfx1250 Tensor Data
// Mover: the [256 x 5] f32 tile is a contiguous 5120 B span, so a 1D TDM
// descriptor streams it linearly into LDS (TENSORcnt-tracked), with
// tensor_dim0 clamped so tail-block OOB elements read as zero.

#define HORIZON 20
#define DT      0.1f
#define EPS     1e-4f
#define V_MAX   0.22f
#define W_MAX   2.8f
#define BLK     256

typedef __attribute__((ext_vector_type(4))) unsigned int u32x4;
typedef __attribute__((ext_vector_type(8))) unsigned int u32x8;

__device__ __forceinline__ float softplusf(float x) {
  // numerically stable log(1+exp(x))
  return fmaxf(x, 0.0f) + log1pf(expf(-fabsf(x)));
}

// Prologue: batch-uniform diagonals from raw params into d_ws[0..7].
__global__ void lqr_consts_kernel(const float* __restrict__ q_raw,
                                  const float* __restrict__ r_raw,
                                  const float* __restrict__ qf_raw,
                                  float* __restrict__ c) {
  if (threadIdx.x == 0) {
    c[0] = softplusf(q_raw[0]) + EPS;          // Q diag
    c[1] = softplusf(q_raw[1]) + EPS;
    c[2] = softplusf(q_raw[2]) + EPS;
    c[3] = softplusf(r_raw[0]) + EPS + EPS;    // R diag + the EPS*I added to S
    c[4] = softplusf(r_raw[1]) + EPS + EPS;
    c[5] = softplusf(qf_raw[0]) + EPS;         // P0 diag
    c[6] = softplusf(qf_raw[1]) + EPS;
    c[7] = softplusf(qf_raw[2]) + EPS;
  }
}

__global__ __launch_bounds__(BLK)
void lqr_kernel(const float* __restrict__ ref,
                const float* __restrict__ c,
                float* __restrict__ out,
                int n) {
  __shared__ __align__(16) float tile[BLK * 5];

  const int t    = threadIdx.x;
  const int row0 = blockIdx.x * BLK;

  // ---- Stage the block's contiguous 5120 B tile into LDS with the TDM ----
  // 1D descriptor: tile_dim0 = 1280 f32 elements; tensor_dim0 clamped to the
  // elements actually remaining so the tail block reads zeros past the end.
  if (t < 32) {  // wave 0 only: TDM ignores EXEC, one DMA per block
    unsigned           lds_off = (unsigned)(size_t)(void*)&tile[0]; // low 32b of LDS generic addr
    unsigned long long ga      = (unsigned long long)(const void*)(ref + (size_t)row0 * 5);
    unsigned           elems   = 5u * (unsigned)(n - row0);   // f32 elements left in tensor

    u32x4 g0;
    g0.x = 1u;                                            // count=1, user mode
    g0.y = lds_off;                                       // lds_addr (bytes)
    g0.z = (unsigned)(ga & 0xFFFFFFFFull);                // global_addr[31:0]
    g0.w = (unsigned)((ga >> 32) & 0x1FFFFFFull)          // global_addr[56:32]
         | (2u << 30);                                    // type=2 ("image")

    u32x8 g1;
    g1.s0 = 2u << 16;                                     // data_size=4B, wg_mask=0
    g1.s1 = (elems & 0xFFFFu) << 16;                      // tensor_dim0[15:0]
    g1.s2 = ((elems >> 16) & 0xFFFFu) | (1u << 16);       // tensor_dim0 hi | tensor_dim1=1
    g1.s3 = (unsigned)(BLK * 5) << 16;                    // tensor_dim1 hi=0 | tile_dim0=1280
    g1.s4 = 0u;                                           // tile_dim1=0 (1D), tile_dim2=0
    g1.s5 = (unsigned)(BLK * 5);                          // tensor_dim0_stride (unused in 1D)
    g1.s6 = 0u;
    g1.s7 = 0u;

    asm volatile("tensor_load_to_lds %0, %1" :: "s"(g0), "s"(g1) : "memory");
    __builtin_amdgcn_s_wait_tensorcnt(0);
  }
  __syncthreads();

  const float e0 = tile[t * 5 + 0];
  const float e1 = tile[t * 5 + 1];
  const float e2 = tile[t * 5 + 2];
  const float vr = tile[t * 5 + 3];
  const float wr = tile[t * 5 + 4];

  // Batch-uniform constants (scalar loads)
  const float q0 = c[0], q1 = c[1], q2 = c[2];
  const float r0e = c[3], r1e = c[4];

  float p00 = c[5], p01 = 0.f, p02 = 0.f, p11 = c[6], p12 = 0.f, p22 = c[7];

  const float dt  = DT;
  const float dt2 = DT * DT;
  const float a   = dt * wr;   // A = [[1,a,0],[-a,1,b],[0,0,1]]
  const float b   = dt * vr;

  float k00 = 0.f, k01 = 0.f, k02 = 0.f, k10 = 0.f, k11 = 0.f, k12 = 0.f;

#pragma unroll
  for (int it = 0; it < HORIZON; ++it) {
    // S = R + eps*I + dt^2 * [[p00,p02],[p02,p22]]  (2x2 symmetric)
    const float s00 = fmaf(dt2, p00, r0e);
    const float s01 = dt2 * p02;
    const float s11 = fmaf(dt2, p22, r1e);
    const float det = fmaf(s00, s11, -(s01 * s01));
    const float inv = 1.0f / det;
    const float i00 =  s11 * inv;
    const float i01 = -s01 * inv;
    const float i11 =  s00 * inv;

    // T = (B^T P) A, B^T P = -dt * [P row0; P row2]
    const float t00 = -dt * fmaf(-a, p01, p00);
    const float t01 = -dt * fmaf( a, p00, p01);
    const float t02 = -dt * fmaf( b, p01, p02);
    const float t10 = -dt * fmaf(-a, p12, p02);
    const float t11 = -dt * fmaf( a, p02, p12);
    const float t12 = -dt * fmaf( b, p12, p22);

    // K = S^-1 T
    k00 = fmaf(i00, t00, i01 * t10);
    k01 = fmaf(i00, t01, i01 * t11);
    k02 = fmaf(i00, t02, i01 * t12);
    k10 = fmaf(i01, t00, i11 * t10);
    k11 = fmaf(i01, t01, i11 * t11);
    k12 = fmaf(i01, t02, i11 * t12);

    // G = A - B K : row0 = A0 + dt*K0, row1 = (-a,1,b), row2 = A2 + dt*K1
    const float g00 = fmaf(dt, k00, 1.f);
    const float g01 = fmaf(dt, k01, a);
    const float g02 = dt * k02;
    const float g20 = dt * k10;
    const float g21 = dt * k11;
    const float g22 = fmaf(dt, k12, 1.f);

    // M = P G (P symmetric; only the entries feeding P_new)
    const float m00 = fmaf(p00, g00, fmaf(-a, p01, p02 * g20));
    const float m01 = fmaf(p00, g01, fmaf(p02, g21, p01));
    const float m02 = fmaf(p00, g02, fmaf(b, p01, p02 * g22));
    const float m10 = fmaf(p01, g00, fmaf(-a, p11, p12 * g20));
    const float m11 = fmaf(p01, g01, fmaf(p12, g21, p11));
    const float m12 = fmaf(p01, g02, fmaf(b, p11, p12 * g22));
    const float m22 = fmaf(p02, g02, fmaf(b, p12, p22 * g22));

    // P_new = Q + A^T M, A^T = [[1,-a,0],[a,1,0],[0,b,1]]
    p00 = q0 + fmaf(-a, m10, m00);
    p01 =      fmaf(-a, m11, m01);
    p02 =      fmaf(-a, m12, m02);
    p11 = q1 + fmaf( a, m01, m11);
    p12 =      fmaf( a, m02, m12);
    p22 = q2 + fmaf( b, m12, m22);
  }

  // delta_u = -K0 @ error ; clip(u_ref + delta_u)
  const float du0 = -fmaf(k00, e0, fmaf(k01, e1, k02 * e2));
  const float du1 = -fmaf(k10, e0, fmaf(k11, e1, k12 * e2));
  const float v = fminf(fmaxf(vr + du0, -V_MAX), V_MAX);
  const float w = fminf(fmaxf(wr + du1, -W_MAX), W_MAX);

  const int row = row0 + t;
  if (row < n) {
    float2 o;
    o.x = v;
    o.y = w;
    *(float2*)(out + 2 * (size_t)row) = o;   // coalesced b64 store
  }
}

extern "C" void kernel_launch(void* const* d_in, const int* in_sizes, int n_in,
                              void* d_out, int out_size, void* d_ws, size_t ws_size,
                              hipStream_t stream) {
  const float* ref    = (const float*)d_in[0];
  const float* q_raw  = (const float*)d_in[1];
  const float* r_raw  = (const float*)d_in[2];
  const float* qf_raw = (const float*)d_in[3];
  float* out = (float*)d_out;
  float* c   = (float*)d_ws;   // 8 floats of uniform constants

  const int n = in_sizes[0] / 5;

  lqr_consts_kernel<<<1, 32, 0, stream>>>(q_raw, r_raw, qf_raw, c);
  lqr_kernel<<<(n + BLK - 1) / BLK, BLK, 0, stream>>>(ref, c, out, n);
}